// NeuraLogicLayer_55628416417928
// MI455X (gfx1250) — compile-verified
//
#include <hip/hip_runtime.h>
#include <stdint.h>

typedef int v4i __attribute__((ext_vector_type(4)));
typedef __attribute__((address_space(3))) const float* lds_cfloat_ptr;

#define BLOCK 256
#define EDGES_PER_THREAD 4
#define MAX_W_LDS 2048   // weight table staged in LDS when W <= this (W = 1024 here)

// ---------------------------------------------------------------------------
// Zero the aggregation buffer and the target-flag buffer.
// ---------------------------------------------------------------------------
__global__ __launch_bounds__(BLOCK) void fill_zero_kernel(float* __restrict__ agg,
                                                          unsigned char* __restrict__ flags,
                                                          int n) {
    int i = blockIdx.x * BLOCK + threadIdx.x;
    if (i < n) {
        agg[i]   = 0.0f;
        flags[i] = 0;
    }
}

// ---------------------------------------------------------------------------
// Mark this layer's target nodes (their old value is zeroed in the combine).
// ---------------------------------------------------------------------------
__global__ __launch_bounds__(BLOCK) void mark_targets_kernel(const int* __restrict__ targets,
                                                             unsigned char* __restrict__ flags,
                                                             int T) {
    int i = blockIdx.x * BLOCK + threadIdx.x;
    if (i < T) flags[targets[i]] = 1;
}

// ---------------------------------------------------------------------------
// Edge scatter: agg[v[e]] += x[u[e]] * weights[widx[e]]
//  - templated on USE_LDS so the LDS weight gather has no global alternative
//    (prevents InstCombine's select-of-pointers -> flat_load merge)
//  - weight table (4KB) staged to LDS via CDNA5 async global->LDS copy;
//    zero-filled first with visible ds_store, barrier-ordered before the
//    async engine's LDS writes
//  - gathers go through a laundered address_space(3) pointer: guaranteed
//    ds_load_b32, and the compiler cannot forward the zeros across the copy
//  - index streams read with non-temporal 128-bit loads (keep x/agg in L2)
//  - scatter via native FP32 global atomics (non-returning)
// ---------------------------------------------------------------------------
template <bool USE_LDS>
__global__ __launch_bounds__(BLOCK) void edge_scatter_kernel(const float* __restrict__ x,
                                                             const float* __restrict__ weights,
                                                             const int* __restrict__ u,
                                                             const int* __restrict__ v,
                                                             const int* __restrict__ widx,
                                                             float* __restrict__ agg,
                                                             long long E, int W) {
    __shared__ __align__(16) float w_lds[MAX_W_LDS];

    lds_cfloat_ptr wp = nullptr;
    if (USE_LDS) {
        // 1) compiler-visible zero-init so w_lds is "stored-to"
        for (int i = threadIdx.x; i < W; i += BLOCK) w_lds[i] = 0.0f;
        __syncthreads();   // DS zeros complete before async engine writes LDS

        // 2) CDNA5 async copy weights -> LDS, one dword per lane.
        //    (GLOBAL_LOAD_ASYNC_TO_LDS_B32; tracked by ASYNCcnt.)
        for (int i = threadIdx.x; i < W; i += BLOCK) {
            unsigned lds_off = (unsigned)(uintptr_t)(&w_lds[i]);   // low 32b of flat = LDS offset
            unsigned long long gaddr = (unsigned long long)(uintptr_t)(weights + i);
            asm volatile("global_load_async_to_lds_b32 %0, %1, off"
                         :: "v"(lds_off), "v"(gaddr) : "memory");
        }
        asm volatile("s_wait_asynccnt 0" ::: "memory");
        __syncthreads();

        // 3) laundered AS(3) pointer: opaque provenance, forces ds_load_b32
        unsigned w_base = (unsigned)(uintptr_t)(&w_lds[0]);
        asm volatile("" : "+s"(w_base));
        wp = reinterpret_cast<lds_cfloat_ptr>(w_base);
    }

    long long t    = (long long)blockIdx.x * BLOCK + threadIdx.x;
    long long base = t * EDGES_PER_THREAD;

    if (base + EDGES_PER_THREAD <= E) {
        // Streaming index loads: non-temporal so 192MB/layer of indices
        // doesn't evict the L2-resident x/agg arrays.
        v4i uu = __builtin_nontemporal_load((const v4i*)(u    + base));
        v4i vv = __builtin_nontemporal_load((const v4i*)(v    + base));
        v4i ww = __builtin_nontemporal_load((const v4i*)(widx + base));
#pragma unroll
        for (int k = 0; k < EDGES_PER_THREAD; ++k) {
            float wv  = USE_LDS ? wp[ww[k]] : weights[ww[k]];   // ds_load_b32 gather
            float msg = x[uu[k]] * wv;
            unsafeAtomicAdd(&agg[vv[k]], msg);   // -> global_atomic_add_f32 (no return)
        }
    } else if (base < E) {
        for (long long e = base; e < E; ++e) {
            float wv  = USE_LDS ? wp[widx[e]] : weights[widx[e]];
            float msg = x[u[e]] * wv;
            unsafeAtomicAdd(&agg[v[e]], msg);
        }
    }
}

// ---------------------------------------------------------------------------
// Combine: x_out[i] = (flags[i] ? 0 : x_in[i]) + act(agg[i])
// act(0)==0 for both Sum and Tanh, so nodes with no in-edges are correct.
// ---------------------------------------------------------------------------
__global__ __launch_bounds__(BLOCK) void combine_kernel(const float* __restrict__ x_in,
                                                        const float* __restrict__ agg,
                                                        const unsigned char* __restrict__ flags,
                                                        float* __restrict__ x_out,
                                                        int n, int use_tanh) {
    int i = blockIdx.x * BLOCK + threadIdx.x;
    if (i >= n) return;
    float a = agg[i];
    if (use_tanh) a = tanhf(a);
    float xv = flags[i] ? 0.0f : x_in[i];
    x_out[i] = xv + a;
}

// ---------------------------------------------------------------------------
// Host-side orchestration (graph-capture safe: only kernel launches on stream)
// ---------------------------------------------------------------------------
extern "C" void kernel_launch(void* const* d_in, const int* in_sizes, int n_in,
                              void* d_out, int out_size, void* d_ws, size_t ws_size,
                              hipStream_t stream) {
    const float* x0      = (const float*)d_in[0];
    const float* weights = (const float*)d_in[1];
    const int*   u       = (const int*)d_in[2];
    const int*   v       = (const int*)d_in[3];
    const int*   widx    = (const int*)d_in[4];
    const int*   targets = (const int*)d_in[5];

    const int       N = in_sizes[0];          // 1,000,000 nodes
    const int       W = in_sizes[1];          // 1024 weights
    const int       L = 2;                    // layers (reference constant)
    const long long E = (long long)in_sizes[2] / L;  // 16M edges/layer
    const int       T = in_sizes[5] / L;      // 250K targets/layer

    // Workspace layout (256B aligned slices): agg[N] f32, x1[N] f32, flags[N] u8
    size_t slice = (((size_t)N * sizeof(float)) + 255) & ~(size_t)255;
    char* ws = (char*)d_ws;
    float*         agg   = (float*)(ws);
    float*         x1    = (float*)(ws + slice);
    unsigned char* flags = (unsigned char*)(ws + 2 * slice);

    const int node_blocks = (N + BLOCK - 1) / BLOCK;
    const int tgt_blocks  = (T + BLOCK - 1) / BLOCK;
    const long long edge_threads = (E + EDGES_PER_THREAD - 1) / EDGES_PER_THREAD;
    const int edge_blocks = (int)((edge_threads + BLOCK - 1) / BLOCK);

    const float* x_in = x0;
    for (int l = 0; l < L; ++l) {
        float* x_out = (l == L - 1) ? (float*)d_out : x1;

        fill_zero_kernel<<<node_blocks, BLOCK, 0, stream>>>(agg, flags, N);
        mark_targets_kernel<<<tgt_blocks, BLOCK, 0, stream>>>(targets + (size_t)l * T, flags, T);

        const int*   ul = u    + (size_t)l * E;
        const int*   vl = v    + (size_t)l * E;
        const int*   wl = widx + (size_t)l * E;
        if (W <= MAX_W_LDS) {
            edge_scatter_kernel<true><<<edge_blocks, BLOCK, 0, stream>>>(
                x_in, weights, ul, vl, wl, agg, E, W);
        } else {
            edge_scatter_kernel<false><<<edge_blocks, BLOCK, 0, stream>>>(
                x_in, weights, ul, vl, wl, agg, E, W);
        }

        combine_kernel<<<node_blocks, BLOCK, 0, stream>>>(
            x_in, agg, flags, x_out, N, /*use_tanh=*/(l == 1) ? 1 : 0);

        x_in = x_out;
    }
}